// ChamferLoss_16879221473852
// MI455X (gfx1250) — compile-verified
//
#include <hip/hip_runtime.h>

typedef __attribute__((ext_vector_type(2))) float v2f;
typedef __attribute__((ext_vector_type(8))) float v8f;

#define CHAMFER_EPS 1e-12f
#define NPTS 1024
#define NTILES 64   // 1024 / 16

__global__ void chamfer_init_out(float* out) {
    if (threadIdx.x == 0) out[0] = 0.0f;
}

// ds_swizzle xor-shuffle within 16-lane halves (group-of-32 encoding:
// offset[14:10]=xor_mask, [4:0]=and_mask). xor masks <16 never cross halves.
__device__ __forceinline__ float minSwz(float v, const int imm_swz) {
    float o;
    switch (imm_swz) {
        case 1: o = __int_as_float(__builtin_amdgcn_ds_swizzle(__float_as_int(v), (1 << 10) | 0x1f)); break;
        case 2: o = __int_as_float(__builtin_amdgcn_ds_swizzle(__float_as_int(v), (2 << 10) | 0x1f)); break;
        case 4: o = __int_as_float(__builtin_amdgcn_ds_swizzle(__float_as_int(v), (4 << 10) | 0x1f)); break;
        default:o = __int_as_float(__builtin_amdgcn_ds_swizzle(__float_as_int(v), (8 << 10) | 0x1f)); break;
    }
    return fminf(v, o);
}

// One block = (cb pair, pass, strip-group).
// Pass 0: A=P rows vs B=Q cols (min_pq); pass 1 swaps roles (min_qp).
//
// K=0 of the 16x16x4 f32 WMMA is free (reference drops channel 0), so A rows
// are staged as (1, -2a.y, -2a.z, -2a.w) and B cols as (|b|^2, b.y, b.z, b.w):
// one WMMA with C = inline 0 yields D[m][n] = |b_n|^2 - 2 a_m.b_n per tile.
// Running min stays in registers; |a|^2 is added once per row, then
// sqrt(max(.,0)+eps) — sqrt is monotone, so min commutes with it (256K sqrts
// instead of 128M).
__launch_bounds__(256)
__global__ void chamfer_kernel(const float* __restrict__ P,
                               const float* __restrict__ Q,
                               float* __restrict__ out)
{
    const int cb   = blockIdx.x;   // (c,b) pair
    const int pass = blockIdx.y;   // 0: A=P,B=Q   1: A=Q,B=P
    const int sgrp = blockIdx.z;   // strip group (0..3)

    const float* Aptr = (pass ? Q : P) + (size_t)cb * NPTS * 4;
    const float* Bptr = (pass ? P : Q) + (size_t)cb * NPTS * 4;

    __shared__ float4 sA[NPTS];    // (1, -2a.y, -2a.z, -2a.w)
    __shared__ float4 sB[NPTS];    // (|b|^2, b.y, b.z, b.w)
    __shared__ float  sAsq[NPTS];  // |a|^2
    __shared__ float  sRed[256];

    for (int i = threadIdx.x; i < NPTS; i += 256) {
        float4 a = ((const float4*)Aptr)[i];
        sAsq[i]  = a.y * a.y + a.z * a.z + a.w * a.w;
        sA[i]    = make_float4(1.0f, -2.0f * a.y, -2.0f * a.z, -2.0f * a.w);
        float4 b = ((const float4*)Bptr)[i];
        float bsq = b.y * b.y + b.z * b.z + b.w * b.w;
        sB[i]    = make_float4(bsq, b.y, b.z, b.w);
    }
    __syncthreads();

    const int  wave = threadIdx.x >> 5;   // 0..7
    const int  lane = threadIdx.x & 31;
    const int  l16  = lane & 15;
    const int  half = lane >> 4;          // 0: K=0,1   1: K=2,3

    // Per-lane float2 views: each lane reads (x,y) or (z,w) of its point.
    const float2* sA2 = (const float2*)sA;
    const float2* sB2 = (const float2*)sB;
    const int laneOff = l16 * 2 + half;   // loop-invariant index part

    // This wave owns two 16-row strips of A; both share every B fragment.
    const int s0 = sgrp * 16 + wave;
    const int s1 = s0 + 8;

    float2 av0 = sA2[s0 * 32 + laneOff];
    float2 av1 = sA2[s1 * 32 + laneOff];
    v2f af0; af0.x = av0.x; af0.y = av0.y;
    v2f af1; af1.x = av1.x; af1.y = av1.y;

    float m0[8], m1[8];
    #pragma unroll
    for (int r = 0; r < 8; ++r) { m0[r] = 3.0e38f; m1[r] = 3.0e38f; }

    // Software-pipelined B-fragment: load ct+1 (wrapped, branch-free) before
    // consuming ct so the s_wait_dscnt at the consumer covers a load issued
    // 2 WMMAs earlier. The wrap load on the last iteration is harmless.
    float2 bvn = sB2[laneOff];            // ct = 0
    #pragma unroll 4
    for (int ct = 0; ct < NTILES; ++ct) {
        v2f bf; bf.x = bvn.x; bf.y = bvn.y;
        bvn = sB2[((ct + 1) & (NTILES - 1)) * 32 + laneOff];

        v8f c0 = {};   // C = inline 0
        v8f c1 = {};
        c0 = __builtin_amdgcn_wmma_f32_16x16x4_f32(
                 false, af0, false, bf, (short)0, c0, false, false);
        c1 = __builtin_amdgcn_wmma_f32_16x16x4_f32(
                 false, af1, false, bf, (short)0, c1, false, false);

        #pragma unroll
        for (int r = 0; r < 8; ++r) {
            m0[r] = fminf(m0[r], c0[r]);
            m1[r] = fminf(m1[r], c1[r]);
        }
    }

    // Min over the 16 columns each half-wave holds (xor tree stays in-half).
    // C/D layout: lanes 0-15 hold rows M=r, lanes 16-31 hold rows M=r+8.
    #pragma unroll
    for (int r = 0; r < 8; ++r) {
        float v0 = m0[r], v1 = m1[r];
        v0 = minSwz(v0, 1); v1 = minSwz(v1, 1);
        v0 = minSwz(v0, 2); v1 = minSwz(v1, 2);
        v0 = minSwz(v0, 4); v1 = minSwz(v1, 4);
        v0 = minSwz(v0, 8); v1 = minSwz(v1, 8);
        m0[r] = v0; m1[r] = v1;
    }

    float sum = 0.0f;
    if (l16 == 0) {
        int rb0 = s0 * 16 + (half ? 8 : 0);
        int rb1 = s1 * 16 + (half ? 8 : 0);
        #pragma unroll
        for (int r = 0; r < 8; ++r) {
            float d0 = m0[r] + sAsq[rb0 + r];          // + |a|^2
            float d1 = m1[r] + sAsq[rb1 + r];
            sum += sqrtf(fmaxf(d0, 0.0f) + CHAMFER_EPS);
            sum += sqrtf(fmaxf(d1, 0.0f) + CHAMFER_EPS);
        }
    }

    // Block-wide sum, one atomicAdd per block
    sRed[threadIdx.x] = sum;
    __syncthreads();
    for (int s = 128; s > 0; s >>= 1) {
        if (threadIdx.x < (unsigned)s) sRed[threadIdx.x] += sRed[threadIdx.x + s];
        __syncthreads();
    }
    if (threadIdx.x == 0) atomicAdd(out, sRed[0]);
}

extern "C" void kernel_launch(void* const* d_in, const int* in_sizes, int n_in,
                              void* d_out, int out_size, void* d_ws, size_t ws_size,
                              hipStream_t stream) {
    const float* p = (const float*)d_in[0];   // (2, B, 1024, 4) f32
    const float* q = (const float*)d_in[1];   // (2, B, 1024, 4) f32
    float* out = (float*)d_out;               // scalar f32

    const int CB = in_sizes[0] / (NPTS * 4);  // 2*B = 128

    chamfer_init_out<<<1, 32, 0, stream>>>(out);
    dim3 grid(CB, 2, 4);                      // (cb, pass, strip-group)
    chamfer_kernel<<<grid, 256, 0, stream>>>(p, q, out);
}